// SAGE_14491219657408
// MI455X (gfx1250) — compile-verified
//
#include <hip/hip_runtime.h>
#include <math.h>

typedef float v2f __attribute__((ext_vector_type(2)));
typedef float v8f __attribute__((ext_vector_type(8)));

#define D_H 256
#define LN_EPS 1e-5f

// ---------------- zero fill (ws is poisoned; accumulators need 0) -------------
__global__ void k_zero_f32(float* __restrict__ p, long n) {
    long i = (long)blockIdx.x * blockDim.x + threadIdx.x;
    long stride = (long)gridDim.x * blockDim.x;
    for (; i < n; i += stride) p[i] = 0.0f;
}

// ---------------- edge scatter: one wave32 per edge ---------------------------
// accum[dst] += h[src]  (256 floats, lane-strided for coalescing), deg[dst] += 1
__global__ void k_scatter_edges(const float* __restrict__ h,
                                const int* __restrict__ src,
                                const int* __restrict__ dst,
                                float* __restrict__ accum,
                                float* __restrict__ deg, int E) {
    int wave = (int)((blockIdx.x * (long)blockDim.x + threadIdx.x) >> 5);
    int lane = threadIdx.x & 31;
    if (wave >= E) return;
    int s = src[wave];
    int d = dst[wave];
    const float* hs = h + (size_t)s * D_H;
    float* ad = accum + (size_t)d * D_H;
#pragma unroll
    for (int i = 0; i < 8; ++i) {
        int c = lane + i * 32;
        atomicAdd(&ad[c], hs[c]);
    }
    if (lane == 0) atomicAdd(&deg[d], 1.0f);
}

// ------------- mean finalize (in place): a = (a + h_dst) / (deg + 1) ----------
__global__ void k_finalize_mean(float* __restrict__ accum,
                                const float* __restrict__ h_dst,
                                const float* __restrict__ deg, long total) {
    long i = (long)blockIdx.x * blockDim.x + threadIdx.x;
    if (i >= total) return;
    long r = i >> 8;  // D_H == 256
    float inv = 1.0f / (deg[r] + 1.0f);
    accum[i] = (accum[i] + h_dst[i]) * inv;
}

// ---------------- fp32 WMMA GEMM: out[M,N] = A[M,256] * W[256,N] + bias -------
// Block = 256 threads = 8 waves arranged 2 (rows) x 4 (cols) -> 32x64 out tile.
// K swept in 2 LDS-staged chunks of 128; each wave runs V_WMMA_F32_16X16X4_F32.
// A LDS rows padded to 132 floats (bank step 4 -> conflict-free strided reads).
#define KCH 128
#define ALD 132
__global__ __launch_bounds__(256) void k_gemm_wmma_f32(
    const float* __restrict__ A, const float* __restrict__ W,
    const float* __restrict__ bias, float* __restrict__ out, int M, int N) {
    __shared__ float Alds[32 * ALD];      // 16.9 KB
    __shared__ float Blds[KCH * 64];      // 32 KB
    const int tid = threadIdx.x;
    const int rowBase = blockIdx.x * 32;
    const int colBase = blockIdx.y * 64;

    const int wave = tid >> 5, lane = tid & 31;
    const int wr = wave >> 2, wc = wave & 3;   // 2x4 wave grid
    const int half = lane >> 4, l16 = lane & 15;

    v8f c = {};  // 16x16 f32 C/D accumulator: VGPR j = row (wr*16 + half*8 + j)

    for (int kb = 0; kb < D_H; kb += KCH) {
        __syncthreads();
        // stage A chunk: 32 rows x 128 K (float4, coalesced), zero OOB rows
#pragma unroll
        for (int j = 0; j < 4; ++j) {
            int idx = tid + j * 256;          // float4 index over 32*32
            int r = idx >> 5, c4 = idx & 31;
            int grow = rowBase + r;
            float4 v = make_float4(0.f, 0.f, 0.f, 0.f);
            if (grow < M) v = *(const float4*)(A + (size_t)grow * D_H + kb + c4 * 4);
            *(float4*)(&Alds[r * ALD + c4 * 4]) = v;
        }
        // stage W chunk: 128 K x 64 N (float4, coalesced)
#pragma unroll
        for (int j = 0; j < 8; ++j) {
            int idx = tid + j * 256;          // float4 index over 128*16
            int k = idx >> 4, c4 = idx & 15;
            float4 v = *(const float4*)(W + (size_t)(kb + k) * N + colBase + c4 * 4);
            *(float4*)(&Blds[k * 64 + c4 * 4]) = v;
        }
        __syncthreads();

        const float* arow = &Alds[(wr * 16 + l16) * ALD];
        const float* bcol = &Blds[wc * 16 + l16];
#pragma unroll 8
        for (int k0 = 0; k0 < KCH; k0 += 4) {
            // 16x4 f32 A frag: lanes 0-15 carry K=k0,k0+1; lanes 16-31 K=k0+2,k0+3
            v2f a, b;
            a.x = arow[k0 + 2 * half];
            a.y = arow[k0 + 2 * half + 1];
            // 4x16 f32 B frag, mirrored lane split across K
            b.x = bcol[(size_t)(k0 + 2 * half) * 64];
            b.y = bcol[(size_t)(k0 + 2 * half + 1) * 64];
            c = __builtin_amdgcn_wmma_f32_16x16x4_f32(
                false, a, false, b, (short)0, c, false, false);
        }
    }

    // store: VGPR j holds (M = wr*16 + half*8 + j, N = wc*16 + l16)
    int col = colBase + wc * 16 + l16;
    float bv = bias[col];
    int rbase = rowBase + wr * 16 + half * 8;
#pragma unroll
    for (int j = 0; j < 8; ++j) {
        int row = rbase + j;
        if (row < M) out[(size_t)row * N + col] = c[j] + bv;
    }
}

// ---------------- LayerNorm(256) + ReLU: one wave32 per row -------------------
__global__ void k_ln_relu(const float* __restrict__ x, const float* __restrict__ g,
                          const float* __restrict__ b, float* __restrict__ y, int M) {
    int wave = (int)((blockIdx.x * (long)blockDim.x + threadIdx.x) >> 5);
    int lane = threadIdx.x & 31;
    if (wave >= M) return;
    const float* xr = x + (size_t)wave * D_H;
    float v[8];
    float s = 0.0f;
#pragma unroll
    for (int i = 0; i < 8; ++i) { v[i] = xr[lane + 32 * i]; s += v[i]; }
#pragma unroll
    for (int o = 16; o > 0; o >>= 1) s += __shfl_xor(s, o, 32);
    float mu = s * (1.0f / 256.0f);
    float vs = 0.0f;
#pragma unroll
    for (int i = 0; i < 8; ++i) { float d = v[i] - mu; vs += d * d; }
#pragma unroll
    for (int o = 16; o > 0; o >>= 1) vs += __shfl_xor(vs, o, 32);
    float rs = rsqrtf(vs * (1.0f / 256.0f) + LN_EPS);
    float* yr = y + (size_t)wave * D_H;
#pragma unroll
    for (int i = 0; i < 8; ++i) {
        int ci = lane + 32 * i;
        float t = (v[i] - mu) * rs * g[ci] + b[ci];
        yr[ci] = fmaxf(t, 0.0f);
    }
}

extern "C" void kernel_launch(void* const* d_in, const int* in_sizes, int n_in,
                              void* d_out, int out_size, void* d_ws, size_t ws_size,
                              hipStream_t stream) {
    (void)n_in; (void)out_size; (void)ws_size;
    const float* feats = (const float*)d_in[0];
    const float* W0    = (const float*)d_in[1];
    const float* b0    = (const float*)d_in[2];
    const float* W1    = (const float*)d_in[3];
    const float* b1    = (const float*)d_in[4];
    const float* ln_g  = (const float*)d_in[5];
    const float* ln_b  = (const float*)d_in[6];
    const int* src0 = (const int*)d_in[7];
    const int* dst0 = (const int*)d_in[8];
    const int* src1 = (const int*)d_in[9];
    const int* dst1 = (const int*)d_in[10];
    const int E0 = in_sizes[7], E1 = in_sizes[9];
    const int N1 = 50000, N2 = 10000, DOUT = 128;

    // workspace layout (floats)
    float* ws     = (float*)d_ws;
    float* accum0 = ws;                            // N1*256 (becomes h_neigh0)
    float* deg0   = accum0 + (size_t)N1 * D_H;     // N1
    float* h      = deg0 + N1;                     // N1*256 (post LN+ReLU)
    float* accum1 = h + (size_t)N1 * D_H;          // N2*256 (becomes h_neigh1)
    float* deg1   = accum1 + (size_t)N2 * D_H;     // N2

    float* h0_out = (float*)d_out;                 // output 1: N1 x 256
    float* out2   = h0_out + (size_t)N1 * D_H;     // output 2: N2 x 128

    // ---- layer 0 ----
    long z0 = (long)N1 * D_H + N1;                 // accum0 + deg0 contiguous
    k_zero_f32<<<4096, 256, 0, stream>>>(accum0, z0);
    {
        long threads = (long)E0 * 32;
        k_scatter_edges<<<(int)((threads + 255) / 256), 256, 0, stream>>>(
            feats, src0, dst0, accum0, deg0, E0);
    }
    {
        long total = (long)N1 * D_H;
        k_finalize_mean<<<(int)((total + 255) / 256), 256, 0, stream>>>(
            accum0, feats, deg0, total);
    }
    {
        dim3 grid((N1 + 31) / 32, D_H / 64);       // N = 256
        k_gemm_wmma_f32<<<grid, 256, 0, stream>>>(accum0, W0, b0, h0_out, N1, D_H);
    }
    {
        long threads = (long)N1 * 32;
        k_ln_relu<<<(int)((threads + 255) / 256), 256, 0, stream>>>(
            h0_out, ln_g, ln_b, h, N1);
    }

    // ---- layer 1 ----
    long z1 = (long)N2 * D_H + N2;                 // accum1 + deg1 contiguous
    k_zero_f32<<<2048, 256, 0, stream>>>(accum1, z1);
    {
        long threads = (long)E1 * 32;
        k_scatter_edges<<<(int)((threads + 255) / 256), 256, 0, stream>>>(
            h, src1, dst1, accum1, deg1, E1);
    }
    {
        long total = (long)N2 * D_H;
        k_finalize_mean<<<(int)((total + 255) / 256), 256, 0, stream>>>(
            accum1, h, deg1, total);
    }
    {
        dim3 grid((N2 + 31) / 32, DOUT / 64);      // N = 128
        k_gemm_wmma_f32<<<grid, 256, 0, stream>>>(accum1, W1, b1, out2, N2, DOUT);
    }
}